// VolumeRenderer_2456721293536
// MI455X (gfx1250) — compile-verified
//
#include <hip/hip_runtime.h>

#define GRID_RES  128
#define BASIS_DIM 9
#define DATA_DIM  28          // 3*9 SH coeffs + 1 sigma
#define N_RAYS    65536
#define N_STEPS   128
#define STEP_SIZE 0.001f
#define BG        1.0f

#define THREADS   128         // 4 wave32s; 42KB LDS/block -> ~7 blocks/WGP

#if defined(__HIP_DEVICE_COMPILE__) && __has_builtin(__builtin_amdgcn_global_load_async_to_lds_b128)
#define USE_ASYNC_LDS 1
#else
#define USE_ASYNC_LDS 0
#endif

#if USE_ASYNC_LDS
typedef int v4i __attribute__((__vector_size__(16)));
typedef __attribute__((address_space(1))) v4i* gv4i_ptr;   // global int4*
typedef __attribute__((address_space(3))) v4i* lv4i_ptr;   // LDS    int4*

// Copy 112 bytes (7 x b128) global -> this lane's LDS slot, async (ASYNCcnt).
// ISA 10.x: INST_OFFSET is added to BOTH the global and the LDS address.
__device__ __forceinline__ void async_sample_copy(const float* g, void* lds_slot) {
    gv4i_ptr gp = (gv4i_ptr)(uintptr_t)g;
    lv4i_ptr lp = (lv4i_ptr)lds_slot;
    __builtin_amdgcn_global_load_async_to_lds_b128(gp, lp,  0, 0);
    __builtin_amdgcn_global_load_async_to_lds_b128(gp, lp, 16, 0);
    __builtin_amdgcn_global_load_async_to_lds_b128(gp, lp, 32, 0);
    __builtin_amdgcn_global_load_async_to_lds_b128(gp, lp, 48, 0);
    __builtin_amdgcn_global_load_async_to_lds_b128(gp, lp, 64, 0);
    __builtin_amdgcn_global_load_async_to_lds_b128(gp, lp, 80, 0);
    __builtin_amdgcn_global_load_async_to_lds_b128(gp, lp, 96, 0);
}
#endif

__global__ __launch_bounds__(THREADS) void volrend_kernel(
    const float* __restrict__ data,      // [128,128,128,28]
    const float* __restrict__ origins,   // [N_RAYS,3]
    const float* __restrict__ dirs,      // [N_RAYS,3]
    const float* __restrict__ viewdirs,  // [N_RAYS,3]
    float* __restrict__ out)             // [N_RAYS,3]
{
    const int tid = threadIdx.x;
    const int r   = blockIdx.x * blockDim.x + tid;
    if (r >= N_RAYS) return;

    // ---- per-ray setup -------------------------------------------------
    float ox = origins[3*r+0], oy = origins[3*r+1], oz = origins[3*r+2];
    float dx = dirs[3*r+0],    dy = dirs[3*r+1],    dz = dirs[3*r+2];
    float vx = viewdirs[3*r+0], vy = viewdirs[3*r+1], vz = viewdirs[3*r+2];

    {   // normalize dir and viewdir
        float inv = 1.0f / sqrtf(dx*dx + dy*dy + dz*dz);
        dx *= inv; dy *= inv; dz *= inv;
        inv = 1.0f / sqrtf(vx*vx + vy*vy + vz*vz);
        vx *= inv; vy *= inv; vz *= inv;
    }

    // ray / unit-cube intersection (dda_unit)
    const float ivx = 1.0f / (dx + 1e-9f);
    const float ivy = 1.0f / (dy + 1e-9f);
    const float ivz = 1.0f / (dz + 1e-9f);
    const float t1x = -ox * ivx, t1y = -oy * ivy, t1z = -oz * ivz;
    const float t2x = t1x + ivx, t2y = t1y + ivy, t2z = t1z + ivz;
    const float tmin = fmaxf(fmaxf(fmaxf(fminf(t1x, t2x), fminf(t1y, t2y)),
                                   fminf(t1z, t2z)), 0.0f);
    const float tmax = fminf(fminf(fminf(fmaxf(t1x, t2x), fmaxf(t1y, t2y)),
                                   fmaxf(t1z, t2z)), 1.0e9f);
    const float dt      = fmaxf(tmax - tmin, 0.0f) * (1.0f / (float)N_STEPS);
    const float delta_t = dt + STEP_SIZE;

    // SH basis (order-2, 9 terms) of viewdir — constant over the march
    const float xx = vx*vx, yy = vy*vy, zz = vz*vz;
    float sh[9];
    sh[0] =  0.28209479177387814f;
    sh[1] = -0.4886025119029199f * vy;
    sh[2] =  0.4886025119029199f * vz;
    sh[3] = -0.4886025119029199f * vx;
    sh[4] =  1.0925484305920792f * vx * vy;
    sh[5] = -1.0925484305920792f * vy * vz;
    sh[6] =  0.31539156525252005f * (2.0f*zz - xx - yy);
    sh[7] = -1.0925484305920792f * vx * vz;
    sh[8] =  0.5462742152960396f * (xx - yy);

    // sample index for step i (truncating cast + clamp == jnp int32-cast + clip)
    auto step_base = [&](int i) -> int {
        const float t  = fmaf((float)i + 0.5f, dt, tmin);
        const float px = fmaf(t, dx, ox) * (float)GRID_RES;
        const float py = fmaf(t, dy, oy) * (float)GRID_RES;
        const float pz = fmaf(t, dz, oz) * (float)GRID_RES;
        int ix = __float2int_rz(px);
        int iy = __float2int_rz(py);
        int iz = __float2int_rz(pz);
        ix = min(max(ix, 0), GRID_RES - 1);
        iy = min(max(iy, 0), GRID_RES - 1);
        iz = min(max(iz, 0), GRID_RES - 1);
        return ((ix * GRID_RES + iy) * GRID_RES + iz) * DATA_DIM;
    };

    float light = 1.0f;
    float acc0 = 0.0f, acc1 = 0.0f, acc2 = 0.0f;

#if USE_ASYNC_LDS
    // ---- async global->LDS pipeline, 2 steps in flight (gfx1250 ASYNCcnt) ----
    // 3 buffers x THREADS lanes x 112B slot.
    // Stride 112B = 28 words: lane base bank = 4*(7i mod 16); 7 coprime 16 =>
    // lanes 0..15 cover all 64 banks, lanes 16..31 are the second (mandatory)
    // pass of a 512B wave read -> conflict-optimal ds_load_b128.
    // (128B stride would put 16 lanes on the same 4-bank group: 16-way conflict.)
    __shared__ float4 lbuf[3][THREADS][7];

    async_sample_copy(data + step_base(0), &lbuf[0][tid][0]);
    async_sample_copy(data + step_base(1), &lbuf[1][tid][0]);

    int rbuf = 0;   // buffer holding step i
    int wbuf = 2;   // buffer for step i+2
    for (int i = 0; i < N_STEPS; ++i) {
        if (i + 2 < N_STEPS) {
            // recycled buffer was last DS-read at iter i-1: close DS/ASYNC hazard
            asm volatile("s_wait_dscnt 0x0" ::: "memory");
            async_sample_copy(data + step_base(i + 2), &lbuf[wbuf][tid][0]);
            // in-order async completion: <=14 outstanding => step i's 7 done
            asm volatile("s_wait_asynccnt 0xe" ::: "memory");
        } else if (i + 1 < N_STEPS) {   // i == N_STEPS-2
            asm volatile("s_wait_asynccnt 0x7" ::: "memory");
        } else {                        // i == N_STEPS-1
            asm volatile("s_wait_asynccnt 0x0" ::: "memory");
        }

        const float4* ls = (const float4*)&lbuf[rbuf][tid][0];
        float4 c0 = ls[0], c1 = ls[1], c2 = ls[2], c3 = ls[3],
               c4 = ls[4], c5 = ls[5], c6 = ls[6];

        rbuf = (rbuf == 2) ? 0 : rbuf + 1;
        wbuf = (wbuf == 2) ? 0 : wbuf + 1;
#else
    // ---- fallback: register double-buffering ---------------------------
    float4 c0, c1, c2, c3, c4, c5, c6;
    {
        const float4* p = (const float4*)(data + step_base(0));
        c0 = p[0]; c1 = p[1]; c2 = p[2]; c3 = p[3]; c4 = p[4]; c5 = p[5]; c6 = p[6];
    }
    for (int i = 0; i < N_STEPS; ++i) {
        const int nx = (i + 1 < N_STEPS) ? (i + 1) : i;
        const float4* np = (const float4*)(data + step_base(nx));
        float4 n0 = np[0], n1 = np[1], n2 = np[2], n3 = np[3],
               n4 = np[4], n5 = np[5], n6 = np[6];
        if (i + 2 < N_STEPS)
            __builtin_prefetch((const void*)(data + step_base(i + 2)), 0, 1);
#endif

        // unpack current sample into scalar regs
        float v[28];
        *(float4*)&v[0]  = c0; *(float4*)&v[4]  = c1; *(float4*)&v[8]  = c2;
        *(float4*)&v[12] = c3; *(float4*)&v[16] = c4; *(float4*)&v[20] = c5;
        *(float4*)&v[24] = c6;

        const float sigma  = fmaxf(v[27], 0.0f);
        const float att    = __expf(-delta_t * sigma);
        const float weight = light * (1.0f - att);

        float dot[3];
#pragma unroll
        for (int c = 0; c < 3; ++c) {
            float s = 0.0f;
#pragma unroll
            for (int k = 0; k < BASIS_DIM; ++k)
                s = fmaf(sh[k], v[c * BASIS_DIM + k], s);
            dot[c] = s;
        }
        acc0 = fmaf(weight, 1.0f / (1.0f + __expf(-dot[0])), acc0);
        acc1 = fmaf(weight, 1.0f / (1.0f + __expf(-dot[1])), acc1);
        acc2 = fmaf(weight, 1.0f / (1.0f + __expf(-dot[2])), acc2);
        light *= att;

#if !USE_ASYNC_LDS
        c0 = n0; c1 = n1; c2 = n2; c3 = n3; c4 = n4; c5 = n5; c6 = n6;
#endif
    }

    out[3*r+0] = acc0 + BG * light;
    out[3*r+1] = acc1 + BG * light;
    out[3*r+2] = acc2 + BG * light;
}

extern "C" void kernel_launch(void* const* d_in, const int* in_sizes, int n_in,
                              void* d_out, int out_size, void* d_ws, size_t ws_size,
                              hipStream_t stream) {
    (void)in_sizes; (void)n_in; (void)out_size; (void)d_ws; (void)ws_size;
    const float* data     = (const float*)d_in[0];
    const float* origins  = (const float*)d_in[1];
    const float* dirs     = (const float*)d_in[2];
    const float* viewdirs = (const float*)d_in[3];
    float* out            = (float*)d_out;

    const int blocks = (N_RAYS + THREADS - 1) / THREADS;
    volrend_kernel<<<blocks, THREADS, 0, stream>>>(data, origins, dirs, viewdirs, out);
}